// AxialTranslatorRotator_50964081935395
// MI455X (gfx1250) — compile-verified
//
#include <hip/hip_runtime.h>
#include <stdint.h>

// Problem constants (from reference)
#define EDGE  255            // W
#define ADDV  127            // (EDGE-1)/2
#define AXH   382            // H = EDGE + ADDV
#define BATCH 16
#define CHAN  32
#define SDIM  765            // 2*H + 1
#define CHUNK 64             // rows staged per TDM transfer (64*255*4 = 65280 B LDS)

typedef unsigned int v4u __attribute__((ext_vector_type(4)));
typedef int          v4i __attribute__((ext_vector_type(4)));
typedef int          v8i __attribute__((ext_vector_type(8)));

// Detect which TDM builtin arity this toolchain uses: the gfx1250 TDM header
// ships only with the amdgpu-toolchain (6-arg builtin); ROCm 7.2 is 5-arg.
#if defined(__has_include)
#if __has_include(<hip/amd_detail/amd_gfx1250_TDM.h>)
#define TDM_6ARG 1
#endif
#endif

__device__ __forceinline__ void tdm_load_to_lds(v4u g0, v8i g1) {
  v4i z4 = (v4i)0;
#ifdef TDM_6ARG
  v8i z8 = (v8i)0;
  __builtin_amdgcn_tensor_load_to_lds(g0, g1, z4, z4, z8, 0);
#else
  __builtin_amdgcn_tensor_load_to_lds(g0, g1, z4, z4, 0);
#endif
}

__device__ __forceinline__ void tdm_store_from_lds(v4u g0, v8i g1) {
  v4i z4 = (v4i)0;
#ifdef TDM_6ARG
  v8i z8 = (v8i)0;
  __builtin_amdgcn_tensor_store_from_lds(g0, g1, z4, z4, z8, 0);
#else
  __builtin_amdgcn_tensor_store_from_lds(g0, g1, z4, z4, 0);
#endif
}

// D# group 0: count=1 (valid), is_restore=0, gather off, lds_addr, 57-bit
// global byte address, type=2 ("image") in bits [127:126].
__device__ __forceinline__ v4u make_g0(uint32_t lds_addr, uint64_t gaddr) {
  v4u g0;
  g0.x = 1u;
  g0.y = lds_addr;
  g0.z = (uint32_t)gaddr;
  g0.w = (uint32_t)((gaddr >> 32) & 0x01FFFFFFu) | (2u << 30);
  return g0;
}

// D# group 1: workgroup_mask=0 (not in cluster), data_size=2 (4 bytes),
// no padding/iteration/atomic-barrier. dims/tiles/strides in element units.
// 2D tile: tile_dim2 = 0 (unused), tensor_dim1_stride = 0 (unused).
__device__ __forceinline__ v8i make_g1(uint32_t dim0, uint32_t dim1,
                                       uint32_t tile0, uint32_t tile1,
                                       uint64_t stride0) {
  v8i g1;
  g1[0] = (int)(2u << 16);                                          // data_size=4B
  g1[1] = (int)((dim0 & 0xFFFFu) << 16);                            // [63:48] tensor_dim0 lo16
  g1[2] = (int)(((dim0 >> 16) & 0xFFFFu) | ((dim1 & 0xFFFFu) << 16)); // dim0 hi16 | dim1 lo16
  g1[3] = (int)(((dim1 >> 16) & 0xFFFFu) | ((tile0 & 0xFFFFu) << 16)); // dim1 hi16 | tile_dim0
  g1[4] = (int)(tile1 & 0xFFFFu);                                   // tile_dim1 | tile_dim2=0
  g1[5] = (int)(uint32_t)(stride0 & 0xFFFFFFFFu);                   // tensor_dim0_stride lo32
  g1[6] = (int)(uint32_t)((stride0 >> 32) & 0xFFFFu);               // stride0 hi16 | dim1_stride lo16=0
  g1[7] = 0;
  return g1;
}

// One workgroup per (b,c) plane. Wave 0: TDM shifted copy (global -> LDS ->
// global with row stride 255 -> 765). Waves 1..N: zero-fill the complement of
// the copy window (disjoint from TDM stores, so no inter-wave ordering needed).
__global__ void AxialTranslatorRotator_scatter_kernel(
    const float* __restrict__ in, const int* __restrict__ offset,
    float* __restrict__ out) {
  extern __shared__ float lds[];

  const int plane = blockIdx.x;           // 0 .. BATCH*CHAN-1
  const int b = plane / CHAN;

  const int q = offset[b * 2 + 0];
  const int r = offset[b * 2 + 1];
  const int uoff = q - (r >> 1);
  const int cu = EDGE - uoff;             // row shift, in [1, 382]
  const int cv = AXH - r;                 // col shift, in [128, 382]

  const float* inPlane = in + (size_t)plane * (AXH * EDGE);
  float* outPlane = out + (size_t)plane * ((size_t)SDIM * SDIM);

  const int wave = (int)(threadIdx.x >> 5);

  if (wave == 0) {
    // LDS byte offset: flat LDS addresses carry the wave-relative LDS offset
    // in addr[31:0] (ISA 10.2 aperture mapping).
    const uint32_t lds_base = (uint32_t)(uintptr_t)(&lds[0]);
    for (int row0 = 0; row0 < AXH; row0 += CHUNK) {
      const int rows = (AXH - row0 < CHUNK) ? (AXH - row0) : CHUNK;
      const uint64_t src =
          (uint64_t)(uintptr_t)(inPlane + (size_t)row0 * EDGE);
      const uint64_t dst =
          (uint64_t)(uintptr_t)(outPlane + (size_t)(cu + row0) * SDIM + cv);

      // Global -> LDS: contiguous 255-wide rows.
      tdm_load_to_lds(make_g0(lds_base, src),
                      make_g1(EDGE, (uint32_t)rows, EDGE, (uint32_t)rows,
                              (uint64_t)EDGE));
      __builtin_amdgcn_s_wait_tensorcnt(0);   // LDS tile resident

      // LDS -> Global: same tile, destination row stride = SDIM (765).
      tdm_store_from_lds(make_g0(lds_base, dst),
                         make_g1(EDGE, (uint32_t)rows, EDGE, (uint32_t)rows,
                                 (uint64_t)SDIM));
      __builtin_amdgcn_s_wait_tensorcnt(0);   // LDS reusable next chunk
    }
  } else {
    // Zero-fill everything outside the [cu,cu+382) x [cv,cv+255) window.
    const uint32_t total = (uint32_t)SDIM * (uint32_t)SDIM;
    const uint32_t tid = (uint32_t)threadIdx.x - 32u;
    const uint32_t stride = (uint32_t)blockDim.x - 32u;
    for (uint32_t idx = tid; idx < total; idx += stride) {
      const uint32_t u = idx / (uint32_t)SDIM;
      const uint32_t v = idx - u * (uint32_t)SDIM;
      const bool inside = ((uint32_t)(u - (uint32_t)cu) < (uint32_t)AXH) &&
                          ((uint32_t)(v - (uint32_t)cv) < (uint32_t)EDGE);
      if (!inside) outPlane[idx] = 0.0f;
    }
  }
}

// Tail outputs: u_min[16], u_max[16], v_min[16], v_max[16] appended flat
// after the big tensor, in return order, value-cast to the f32 output buffer.
__global__ void AxialTranslatorRotator_bounds_kernel(
    const int* __restrict__ offset, float* __restrict__ out_tail) {
  const int t = (int)threadIdx.x;
  if (t < 4 * BATCH) {
    const int b = t & (BATCH - 1);
    const int which = t >> 4;
    const int q = offset[b * 2 + 0];
    const int r = offset[b * 2 + 1];
    const int uoff = q - (r >> 1);
    const int cu = EDGE - uoff;   // u_min
    const int cv = AXH - r;       // v_min
    int val;
    if (which == 0)      val = cu;          // u_min
    else if (which == 1) val = cu + AXH;    // u_max = u_min + H
    else if (which == 2) val = cv;          // v_min
    else                 val = cv + EDGE;   // v_max = v_min + W
    out_tail[t] = (float)val;
  }
}

extern "C" void kernel_launch(void* const* d_in, const int* in_sizes, int n_in,
                              void* d_out, int out_size, void* d_ws, size_t ws_size,
                              hipStream_t stream) {
  (void)in_sizes; (void)n_in; (void)out_size; (void)d_ws; (void)ws_size;

  const float* in = (const float*)d_in[0];
  const int* offset = (const int*)d_in[1];
  float* out = (float*)d_out;

  const int planes = BATCH * CHAN;                       // 512 workgroups
  const size_t ldsBytes = (size_t)CHUNK * EDGE * sizeof(float);  // 65280 B

  AxialTranslatorRotator_scatter_kernel<<<planes, 512, ldsBytes, stream>>>(
      in, offset, out);

  float* tail = out + (size_t)planes * (size_t)SDIM * (size_t)SDIM;
  AxialTranslatorRotator_bounds_kernel<<<1, 64, 0, stream>>>(offset, tail);
}